// RecurrentDecoder_44813688767176
// MI455X (gfx1250) — compile-verified
//
#include <hip/hip_runtime.h>
#include <cstdint>
#include <cstddef>

// ---------------- problem constants (match reference) ----------------
#define N1C 20000
#define N2C 20000
#define NNC 40000        // N1 + N2
#define EEC 160000
#define E3C 480000       // 3 * E
#define HHC 4            // heads

typedef float v2f __attribute__((ext_vector_type(2)));
typedef float v8f __attribute__((ext_vector_type(8)));

// ---------------- small helpers ----------------
__device__ __forceinline__ float waveReduceSum(float v) {
    #pragma unroll
    for (int off = 16; off > 0; off >>= 1) v += __shfl_xor(v, off, 32);
    return v;
}

__device__ __forceinline__ unsigned floatToKey(float x) {
    unsigned b = __float_as_uint(x);
    return (b & 0x80000000u) ? ~b : (b | 0x80000000u);
}
__device__ __forceinline__ float keyToFloat(unsigned k) {
    unsigned b = (k & 0x80000000u) ? (k & 0x7FFFFFFFu) : ~k;
    return __uint_as_float(b);
}

// ---------------- zero fill ----------------
__global__ void zero_f32_kernel(float* p, size_t n) {
    size_t i = (size_t)blockIdx.x * blockDim.x + threadIdx.x;
    if (i < n) p[i] = 0.0f;
}

// ---------------- concat x = [x1; x2] ----------------
__global__ void concat_kernel(const float* __restrict__ x1,
                              const float* __restrict__ x2,
                              float* __restrict__ feat) {
    size_t i = (size_t)blockIdx.x * blockDim.x + threadIdx.x;
    size_t total = (size_t)NNC * 128;
    if (i >= total) return;
    size_t half = (size_t)N1C * 128;
    feat[i] = (i < half) ? x1[i] : x2[i - half];
}

// ---------------- build merged edge list ----------------
__global__ void build_edges_kernel(const int* __restrict__ ei1,
                                   const int* __restrict__ ei2,
                                   int* __restrict__ src,
                                   int* __restrict__ dst) {
    int j = blockIdx.x * blockDim.x + threadIdx.x;
    if (j >= E3C) return;
    int s, d;
    if (j < EEC)            { s = ei1[j];              d = ei1[EEC + j]; }
    else if (j < 2 * EEC)   { int k = j - EEC;   s = ei2[k] + N1C; d = ei2[EEC + k] + N1C; }
    else                    { int k = j - 2*EEC; s = ei1[k] + N1C; d = ei1[EEC + k] + N1C; }
    src[j] = s; dst[j] = d;
}

// ---------------- fp32 WMMA GEMM: C[M,Nc] = A[M,K] @ B[K,Nc] ----------------
// One wave computes a 16x64 output strip using V_WMMA_F32_16X16X4_F32.
// fp32 A layout 16x4: lanes 0-15 -> M=lane, v0=K0,v1=K1; lanes 16-31 -> v0=K2,v1=K3.
// fp32 B layout 4x16: lanes 0-15 -> N=lane, v0=K0,v1=K1; lanes 16-31 -> v0=K2,v1=K3.
// fp32 C layout 16x16: vgpr v, lanes 0-15 -> M=v, lanes 16-31 -> M=v+8.
__global__ void __launch_bounds__(256)
gemm16x64_wmma_f32(const float* __restrict__ A, const float* __restrict__ B,
                   float* __restrict__ C, int K, int Nc, int nStrips) {
    const int lane = threadIdx.x & 31;
    const int wave = threadIdx.x >> 5;
    int strip = blockIdx.x * 8 + wave;
    if (strip >= nStrips) strip = nStrips - 1;   // wave-uniform clamp: EXEC stays all-ones
    const int stripsPerRow = Nc >> 6;            // Nc / 64
    const int tm = strip / stripsPerRow;
    const int tn = strip - tm * stripsPerRow;
    const int l15 = lane & 15;
    const int hi  = lane >> 4;                   // 0 or 1 (K-half select)

    const float* Ap = A + (size_t)((tm << 4) + l15) * K + (hi << 1);
    const float* Bp = B + (size_t)(hi << 1) * Nc + (tn << 6) + l15;

    v8f acc0 = {}, acc1 = {}, acc2 = {}, acc3 = {};
    for (int k = 0; k < K; k += 4) {
        v2f a;
        a.x = Ap[k];
        a.y = Ap[k + 1];
        const float* Bk = Bp + (size_t)k * Nc;
        v2f b0, b1, b2, b3;
        b0.x = Bk[0];   b0.y = Bk[Nc + 0];
        b1.x = Bk[16];  b1.y = Bk[Nc + 16];
        b2.x = Bk[32];  b2.y = Bk[Nc + 32];
        b3.x = Bk[48];  b3.y = Bk[Nc + 48];
        acc0 = __builtin_amdgcn_wmma_f32_16x16x4_f32(false, a, false, b0, (short)0, acc0, false, false);
        acc1 = __builtin_amdgcn_wmma_f32_16x16x4_f32(false, a, false, b1, (short)0, acc1, false, false);
        acc2 = __builtin_amdgcn_wmma_f32_16x16x4_f32(false, a, false, b2, (short)0, acc2, false, false);
        acc3 = __builtin_amdgcn_wmma_f32_16x16x4_f32(false, a, false, b3, (short)0, acc3, false, false);
    }

    float* Cp = C + (size_t)((tm << 4) + (hi << 3)) * Nc + (tn << 6) + l15;
    #pragma unroll
    for (int v = 0; v < 8; ++v) {
        Cp[(size_t)v * Nc + 0]  = acc0[v];
        Cp[(size_t)v * Nc + 16] = acc1[v];
        Cp[(size_t)v * Nc + 32] = acc2[v];
        Cp[(size_t)v * Nc + 48] = acc3[v];
    }
}

// ---------------- attention scores: a_s[n,h] = <h[n,h,:], att_src[h,:]> ----------------
__global__ void scores_kernel(const float* __restrict__ h,
                              const float* __restrict__ att_s,
                              const float* __restrict__ att_d,
                              float* __restrict__ as_, float* __restrict__ ad_,
                              int C) {
    int w    = (blockIdx.x * blockDim.x + threadIdx.x) >> 5;
    int lane = threadIdx.x & 31;
    if (w >= NNC * HHC) return;
    int n  = w >> 2;
    int hd = w & 3;
    const float* hp  = h + (size_t)n * (HHC * C) + hd * C;
    const float* asp = att_s + hd * C;
    const float* adp = att_d + hd * C;
    float ss = 0.0f, sd = 0.0f;
    for (int c = lane; c < C; c += 32) {
        float v = hp[c];
        ss += v * asp[c];
        sd += v * adp[c];
    }
    ss = waveReduceSum(ss);
    sd = waveReduceSum(sd);
    if (lane == 0) { as_[w] = ss; ad_[w] = sd; }
}

// ---------------- edge pass 1: segment max (monotone-key atomicMax) ----------------
__global__ void edge_max_kernel(const int* __restrict__ src, const int* __restrict__ dst,
                                const float* __restrict__ as_, const float* __restrict__ ad_,
                                unsigned* __restrict__ emaxK) {
    int t = blockIdx.x * blockDim.x + threadIdx.x;
    if (t >= E3C * HHC) return;
    int e  = t >> 2;
    int hd = t & 3;
    int d  = dst[e];
    float x = as_[src[e] * HHC + hd] + ad_[d * HHC + hd];
    x = (x > 0.0f) ? x : 0.2f * x;                         // leaky_relu(0.2)
    atomicMax(&emaxK[d * HHC + hd], floatToKey(x));
}

// ---------------- edge pass 2: exp(e - max), denom sum ----------------
__global__ void edge_exp_kernel(const int* __restrict__ src, const int* __restrict__ dst,
                                const float* __restrict__ as_, const float* __restrict__ ad_,
                                const unsigned* __restrict__ emaxK,
                                float* __restrict__ denom, float* __restrict__ eexp) {
    int t = blockIdx.x * blockDim.x + threadIdx.x;
    if (t >= E3C * HHC) return;
    int e  = t >> 2;
    int hd = t & 3;
    int d  = dst[e];
    float x = as_[src[e] * HHC + hd] + ad_[d * HHC + hd];
    x = (x > 0.0f) ? x : 0.2f * x;
    float m  = keyToFloat(emaxK[d * HHC + hd]);
    float ex = expf(x - m);
    eexp[t] = ex;
    atomicAdd(&denom[d * HHC + hd], ex);
}

// ---------------- edge pass 3: scatter-add alpha * h[src] ----------------
__global__ void edge_agg_kernel(const int* __restrict__ src, const int* __restrict__ dst,
                                const float* __restrict__ eexp, const float* __restrict__ denom,
                                const float* __restrict__ h, float* __restrict__ agg, int C) {
    long long w = ((long long)blockIdx.x * blockDim.x + threadIdx.x) >> 5;
    int lane = threadIdx.x & 31;
    if (w >= (long long)E3C * HHC) return;
    int e  = (int)(w >> 2);
    int hd = (int)(w & 3);
    int d = dst[e], s = src[e];
    float alpha = eexp[w] / (denom[d * HHC + hd] + 1e-16f);
    const float* hp = h   + (size_t)s * (HHC * C) + hd * C;
    float*       ap = agg + (size_t)d * (HHC * C) + hd * C;
    for (int c = lane; c < C; c += 32)
        atomicAdd(&ap[c], alpha * hp[c]);
}

// ---------------- mean over heads + bias (+ optional relu) ----------------
__global__ void finalize_kernel(const float* __restrict__ agg, const float* __restrict__ bias,
                                float* __restrict__ feat, int C, int relu) {
    size_t t = (size_t)blockIdx.x * blockDim.x + threadIdx.x;
    if (t >= (size_t)NNC * C) return;
    int n = (int)(t / C);
    int c = (int)(t - (size_t)n * C);
    const float* ap = agg + (size_t)n * (HHC * C) + c;
    float v = 0.25f * (ap[0] + ap[C] + ap[2 * C] + ap[3 * C]) + bias[c];
    if (relu) v = fmaxf(v, 0.0f);
    feat[t] = v;
}

// ---------------- final: row-normalize h[n1:] and add residual ----------------
__global__ void final_kernel(const float* __restrict__ hf, const float* __restrict__ res,
                             const float* __restrict__ bres, float* __restrict__ out) {
    int w    = (blockIdx.x * blockDim.x + threadIdx.x) >> 5;
    int lane = threadIdx.x & 31;
    if (w >= N2C) return;
    const float* row = hf + (size_t)(N1C + w) * 128;
    float vals[4];
    float ss = 0.0f;
    #pragma unroll
    for (int i = 0; i < 4; ++i) {
        vals[i] = row[lane + 32 * i];
        ss += vals[i] * vals[i];
    }
    ss = waveReduceSum(ss);
    float inv = 1.0f / fmaxf(sqrtf(ss), 1e-12f);
    #pragma unroll
    for (int i = 0; i < 4; ++i) {
        int c = lane + 32 * i;
        out[(size_t)w * 128 + c] = vals[i] * inv + res[(size_t)w * 128 + c] + bres[c];
    }
}

// ---------------- host driver ----------------
extern "C" void kernel_launch(void* const* d_in, const int* in_sizes, int n_in,
                              void* d_out, int out_size, void* d_ws, size_t ws_size,
                              hipStream_t stream) {
    (void)in_sizes; (void)n_in; (void)out_size; (void)ws_size;

    const float* x1   = (const float*)d_in[0];
    const float* x2   = (const float*)d_in[1];
    const int*   ei1  = (const int*)d_in[2];
    const int*   ei2  = (const int*)d_in[3];
    const float* W0   = (const float*)d_in[4];
    const float* a0s  = (const float*)d_in[5];
    const float* a0d  = (const float*)d_in[6];
    const float* b0   = (const float*)d_in[7];
    const float* W1   = (const float*)d_in[8];
    const float* a1s  = (const float*)d_in[9];
    const float* a1d  = (const float*)d_in[10];
    const float* b1   = (const float*)d_in[11];
    const float* W2   = (const float*)d_in[12];
    const float* a2s  = (const float*)d_in[13];
    const float* a2d  = (const float*)d_in[14];
    const float* b2   = (const float*)d_in[15];
    const float* Wf   = (const float*)d_in[16];
    const float* afs  = (const float*)d_in[17];
    const float* afd  = (const float*)d_in[18];
    const float* bf   = (const float*)d_in[19];
    const float* Wres = (const float*)d_in[20];
    const float* bres = (const float*)d_in[21];
    float* out = (float*)d_out;

    // workspace carve-up (~208 MB)
    char* base = (char*)d_ws;
    size_t off = 0;
    auto carve = [&](size_t bytes) -> char* {
        char* p = base + off;
        off += (bytes + 255) & ~(size_t)255;
        return p;
    };
    float*    feat  = (float*)   carve((size_t)NNC * 128 * 4);
    float*    h     = (float*)   carve((size_t)NNC * 512 * 4);
    float*    agg   = (float*)   carve((size_t)NNC * 512 * 4);
    float*    as_   = (float*)   carve((size_t)NNC * HHC * 4);
    float*    ad_   = (float*)   carve((size_t)NNC * HHC * 4);
    unsigned* emaxK = (unsigned*)carve((size_t)NNC * HHC * 4);
    float*    denom = (float*)   carve((size_t)NNC * HHC * 4);
    float*    eexp  = (float*)   carve((size_t)E3C * HHC * 4);
    int*      srcA  = (int*)     carve((size_t)E3C * 4);
    int*      dstA  = (int*)     carve((size_t)E3C * 4);
    float*    res   = (float*)   carve((size_t)N2C * 128 * 4);

    // x = concat(x1, x2)
    {
        size_t n = (size_t)NNC * 128;
        concat_kernel<<<(unsigned)((n + 255) / 256), 256, 0, stream>>>(x1, x2, feat);
    }
    // merged edge list
    build_edges_kernel<<<(E3C + 255) / 256, 256, 0, stream>>>(ei1, ei2, srcA, dstA);
    // residual = x2 @ Wres   (bres added in final kernel)
    {
        int strips = (N2C / 16) * (128 / 64);
        gemm16x64_wmma_f32<<<(strips + 7) / 8, 256, 0, stream>>>(x2, Wres, res, 128, 128, strips);
    }

    struct LayerCfg { const float *W, *as, *ad, *b; int Cin, Cout, relu; };
    LayerCfg Ls[4] = {
        { W0, a0s, a0d, b0, 128,  64, 1 },
        { W1, a1s, a1d, b1,  64,  64, 1 },
        { W2, a2s, a2d, b2,  64, 128, 1 },
        { Wf, afs, afd, bf, 128, 128, 0 },
    };

    for (int li = 0; li < 4; ++li) {
        const LayerCfg& L = Ls[li];
        const int HC = HHC * L.Cout;

        // h = feat @ W   (N x HC)  -- fp32 WMMA
        {
            int strips = (NNC / 16) * (HC / 64);
            gemm16x64_wmma_f32<<<(strips + 7) / 8, 256, 0, stream>>>(feat, L.W, h, L.Cin, HC, strips);
        }
        // per-(node, head) attention scalars
        {
            int waves = NNC * HHC;
            scores_kernel<<<(waves * 32 + 255) / 256, 256, 0, stream>>>(h, L.as, L.ad, as_, ad_, L.Cout);
        }
        // reset accumulators (key 0 == identity for the monotone max key)
        zero_f32_kernel<<<(NNC * HHC + 255) / 256, 256, 0, stream>>>((float*)emaxK, (size_t)NNC * HHC);
        zero_f32_kernel<<<(NNC * HHC + 255) / 256, 256, 0, stream>>>(denom, (size_t)NNC * HHC);
        {
            size_t n = (size_t)NNC * HC;
            zero_f32_kernel<<<(unsigned)((n + 255) / 256), 256, 0, stream>>>(agg, n);
        }
        // edge softmax + aggregate
        edge_max_kernel<<<(E3C * HHC + 255) / 256, 256, 0, stream>>>(srcA, dstA, as_, ad_, emaxK);
        edge_exp_kernel<<<(E3C * HHC + 255) / 256, 256, 0, stream>>>(srcA, dstA, as_, ad_, emaxK, denom, eexp);
        {
            long long waves = (long long)E3C * HHC;
            long long thr = waves * 32;
            edge_agg_kernel<<<(unsigned)((thr + 255) / 256), 256, 0, stream>>>(srcA, dstA, eexp, denom, h, agg, L.Cout);
        }
        // mean over heads + bias (+relu), write back into feat for next layer
        {
            size_t n = (size_t)NNC * L.Cout;
            finalize_kernel<<<(unsigned)((n + 255) / 256), 256, 0, stream>>>(agg, L.b, feat, L.Cout, L.relu);
        }
    }

    // row-normalize tail + residual
    final_kernel<<<(N2C * 32 + 255) / 256, 256, 0, stream>>>(feat, res, bres, out);
}